// KWinnerLayer_57088705298665
// MI455X (gfx1250) — compile-verified
//
#include <hip/hip_runtime.h>

typedef __attribute__((ext_vector_type(2))) float v2f;
typedef __attribute__((ext_vector_type(8))) float v8f;

#define BATCH 8192
#define FEATS 4096
#define RANK_LOW 211   // k = 4096*0.05 + 7
#define RANK_HIGH 7    // TOP_IGNORE
#define DENSITY 0.05f
#define EPT 16         // elements per thread in kernel C (FEATS / 256)

// ---------------------------------------------------------------------------
// Kernel A: factor[j] = exp((DENSITY - active_average[j]) * BOOST_FACTOR)
// ---------------------------------------------------------------------------
__global__ void kw_factor_kernel(const float* __restrict__ aa,
                                 float* __restrict__ factor) {
    int j = blockIdx.x * blockDim.x + threadIdx.x;
    if (j < FEATS) {
        factor[j] = __expf((DENSITY - aa[j]) * 1.0f);
    }
}

// ---------------------------------------------------------------------------
// Kernel B: col_mean via WMMA.  col_sum = ones(1,B) @ x(B,F).
// A (16x4) = all ones, B-tile = x[r..r+3, col0..col0+15]  ->  every row of the
// 16x16 f32 accumulator D holds the running column sums of the tile.
// One workgroup (8 waves, wave32) per 16-column tile; wave w sums rows
// [w*1024, (w+1)*1024); deterministic LDS reduction across waves.
// ---------------------------------------------------------------------------
__global__ void kw_colmean_wmma_kernel(const float* __restrict__ x,
                                       float* __restrict__ colmean) {
    const int col0 = blockIdx.x * 16;
    const int lane = threadIdx.x & 31;
    const int wave = threadIdx.x >> 5;            // 0..7
    const int n    = lane & 15;                   // column within tile
    const int kb   = (lane >> 4) * 2;             // 0 (lanes 0-15) / 2 (16-31)

    v2f a; a.x = 1.0f; a.y = 1.0f;                // ones A-matrix (16x4)
    v8f c = {};                                   // f32 accumulator

    const float* base = x + col0 + n;
    const int r0 = wave * (BATCH / 8);
    const int r1 = r0 + (BATCH / 8);
    for (int r = r0; r < r1; r += 4) {
        v2f b;
        b.x = base[(size_t)(r + kb) * FEATS];     // K = kb   row
        b.y = base[(size_t)(r + kb + 1) * FEATS]; // K = kb+1 row
        c = __builtin_amdgcn_wmma_f32_16x16x4_f32(
                false, a, false, b, (short)0, c, false, false);
    }

    __shared__ float part[8][16];
    if (lane < 16) part[wave][lane] = c[0];       // row M=0 of D = col sums
    __syncthreads();
    if (threadIdx.x < 16) {
        float s = 0.0f;
        for (int w = 0; w < 8; ++w) s += part[w][threadIdx.x];
        colmean[col0 + threadIdx.x] = s * (1.0f / (float)BATCH);
    }
}

// ---------------------------------------------------------------------------
// Kernel C: per-row radix-select thresholds + masked output.
// Keys and x stay in VGPRs; only the 256-bin histogram lives in LDS.
// ---------------------------------------------------------------------------
__device__ __forceinline__ unsigned order_key(float f) {
    unsigned u = __float_as_uint(f);
    return (u & 0x80000000u) ? ~u : (u | 0x80000000u);   // ascending-order map
}

// Returns the full 32-bit key of the R-th largest element among the FEATS keys
// distributed as rk[0..15] per thread.  4 passes x 8 bits; LDS histogram.
__device__ __forceinline__ unsigned radix_select_desc(
        const unsigned rk[EPT], unsigned* __restrict__ hist,
        unsigned* __restrict__ s_prefix, int* __restrict__ s_target,
        int R, int t) {
    unsigned prefix = 0;
    int target = R;
    for (int shift = 24; shift >= 0; shift -= 8) {
        hist[t] = 0;
        __syncthreads();
        const unsigned hmask = (shift == 24) ? 0u : (0xFFFFFFFFu << (shift + 8));
        #pragma unroll
        for (int e = 0; e < EPT; ++e) {
            unsigned k = rk[e];
            if ((k & hmask) == prefix)
                atomicAdd(&hist[(k >> shift) & 0xFFu], 1u);
        }
        __syncthreads();
        if (t == 0) {
            int tgt = target;
            int b = 255;
            for (; b > 0; --b) {
                int cnt = (int)hist[b];
                if (tgt <= cnt) break;
                tgt -= cnt;
            }
            *s_prefix = prefix | ((unsigned)b << shift);
            *s_target = tgt;
        }
        __syncthreads();
        prefix = *s_prefix;
        target = *s_target;
    }
    return prefix;   // exact key of the R-th largest value
}

__global__ void __launch_bounds__(256)
kw_row_kernel(const float* __restrict__ x,
              const float* __restrict__ factor,
              const float* __restrict__ colmean,
              float* __restrict__ out) {
    __shared__ unsigned hist[256];
    __shared__ unsigned s_prefix;
    __shared__ int      s_target;

    const int row = blockIdx.x;
    const int t   = threadIdx.x;          // 0..255 (8 waves, wave32)

    const float4* __restrict__ xin = (const float4*)(x + (size_t)row * FEATS);
    const float4* __restrict__ fin = (const float4*)factor;

    float    rx[EPT];                     // raw x values (registers)
    unsigned rk[EPT];                     // order keys of xp (registers)

    // Load row once (coalesced b128), build order keys of xp = x * factor.
    #pragma unroll
    for (int it = 0; it < EPT / 4; ++it) {
        int idx = it * 256 + t;           // float4 index
        float4 xv = xin[idx];
        float4 fv = fin[idx];
        rx[it * 4 + 0] = xv.x;  rk[it * 4 + 0] = order_key(xv.x * fv.x);
        rx[it * 4 + 1] = xv.y;  rk[it * 4 + 1] = order_key(xv.y * fv.y);
        rx[it * 4 + 2] = xv.z;  rk[it * 4 + 2] = order_key(xv.z * fv.z);
        rx[it * 4 + 3] = xv.w;  rk[it * 4 + 3] = order_key(xv.w * fv.w);
    }

    const unsigned thrHigh = radix_select_desc(rk, hist, &s_prefix, &s_target,
                                               RANK_HIGH, t);   // 7th largest
    const unsigned thrLow  = radix_select_desc(rk, hist, &s_prefix, &s_target,
                                               RANK_LOW, t);    // 211th largest

    // out = x * (thrLow <= xp <= thrHigh) - col_mean * (xp <= thrHigh)
    const float4* __restrict__ cmin = (const float4*)colmean;
    float4* __restrict__ oout = (float4*)(out + (size_t)row * FEATS);
    #pragma unroll
    for (int it = 0; it < EPT / 4; ++it) {
        int idx = it * 256 + t;
        float4 cm = cmin[idx];
        float4 o;
        {
            unsigned k = rk[it * 4 + 0];
            bool m2 = (k <= thrHigh);
            bool m  = m2 && (k >= thrLow);
            o.x = (m ? rx[it * 4 + 0] : 0.0f) - (m2 ? cm.x : 0.0f);
        }
        {
            unsigned k = rk[it * 4 + 1];
            bool m2 = (k <= thrHigh);
            bool m  = m2 && (k >= thrLow);
            o.y = (m ? rx[it * 4 + 1] : 0.0f) - (m2 ? cm.y : 0.0f);
        }
        {
            unsigned k = rk[it * 4 + 2];
            bool m2 = (k <= thrHigh);
            bool m  = m2 && (k >= thrLow);
            o.z = (m ? rx[it * 4 + 2] : 0.0f) - (m2 ? cm.z : 0.0f);
        }
        {
            unsigned k = rk[it * 4 + 3];
            bool m2 = (k <= thrHigh);
            bool m  = m2 && (k >= thrLow);
            o.w = (m ? rx[it * 4 + 3] : 0.0f) - (m2 ? cm.w : 0.0f);
        }
        oout[idx] = o;
    }
}

// ---------------------------------------------------------------------------
// Launcher
// ---------------------------------------------------------------------------
extern "C" void kernel_launch(void* const* d_in, const int* in_sizes, int n_in,
                              void* d_out, int out_size, void* d_ws, size_t ws_size,
                              hipStream_t stream) {
    const float* x  = (const float*)d_in[0];           // (8192, 4096) f32
    const float* aa = (const float*)d_in[1];           // (4096,) f32
    float* out = (float*)d_out;                        // (8192, 4096) f32

    float* factor  = (float*)d_ws;                     // 4096 f32
    float* colmean = (float*)d_ws + FEATS;             // 4096 f32

    // A: per-column boost factor
    kw_factor_kernel<<<FEATS / 256, 256, 0, stream>>>(aa, factor);

    // B: column means via WMMA ones-GEMV (one block per 16-column tile)
    kw_colmean_wmma_kernel<<<FEATS / 16, 256, 0, stream>>>(x, colmean);

    // C: per-row radix select + masked output (one block per row)
    kw_row_kernel<<<BATCH, 256, 0, stream>>>(x, factor, colmean, out);
}